// TemporalGCN_56075093016576
// MI455X (gfx1250) — compile-verified
//
#include <hip/hip_runtime.h>

typedef __attribute__((ext_vector_type(2))) float v2f;
typedef __attribute__((ext_vector_type(8))) float v8f;

#define HID 128
#define IN_CH 16
#define T_STEPS 12
#define PRED 6

// ---------------- degree / normalization ----------------
__global__ void k_init_deg(float* deg, int n) {
    int i = blockIdx.x * blockDim.x + threadIdx.x;
    if (i < n) deg[i] = 1.0f;  // self loop
}

__global__ void k_acc_deg(const int* __restrict__ dst, const float* __restrict__ ew,
                          float* __restrict__ deg, int E) {
    int e = blockIdx.x * blockDim.x + threadIdx.x;
    if (e < E) atomicAdd(&deg[dst[e]], ew[e]);
}

__global__ void k_rsqrt(float* deg, int n) {
    int i = blockIdx.x * blockDim.x + threadIdx.x;
    if (i < n) deg[i] = rsqrtf(deg[i]);
}

// ---------------- dense GEMM via native f32 WMMA ----------------
// H[BN,128] = A[BN,Fin] @ W[Fin,128]   (bias added later in finalize)
// One block = 8 waves; wave w computes a 16x16 tile at (blockIdx.x*16, w*16).
__global__ void __launch_bounds__(256) k_gemm_wmma(const float* __restrict__ A,
                                                   const float* __restrict__ W,
                                                   float* __restrict__ H, int Fin) {
    const int lane = threadIdx.x & 31;
    const int wave = threadIdx.x >> 5;   // column tile 0..7 (HID/16)
    const int hi   = lane >> 4;          // K-half select (0/1)
    const int ml   = lane & 15;
    const int row  = blockIdx.x * 16 + ml;
    const int col  = wave * 16 + ml;

    const float* arow = A + (size_t)row * Fin;
    v8f c = {0.f, 0.f, 0.f, 0.f, 0.f, 0.f, 0.f, 0.f};

    for (int k = 0; k < Fin; k += 4) {
        // A 16x4 f32 layout: lanes 0-15 hold K={k,k+1}, lanes 16-31 K={k+2,k+3}
        v2f a;
        a.x = arow[k + 2 * hi];
        a.y = arow[k + 2 * hi + 1];
        // B 4x16 f32 (row-major W[Fin,128]): mirrored K-half layout, N across lanes
        v2f b;
        b.x = W[(size_t)(k + 2 * hi) * HID + col];
        b.y = W[(size_t)(k + 2 * hi + 1) * HID + col];
        c = __builtin_amdgcn_wmma_f32_16x16x4_f32(
                /*neg_a=*/false, a, /*neg_b=*/false, b,
                /*c_mod=*/(short)0, c, /*reuse_a=*/false, /*reuse_b=*/false);
    }

    // D layout: VGPR r -> row r (lanes 0-15) / row r+8 (lanes 16-31)
    float* hbase = H + ((size_t)blockIdx.x * 16 + 8 * hi) * HID + wave * 16 + ml;
#pragma unroll
    for (int r = 0; r < 8; ++r)
        hbase[(size_t)r * HID] = c[r];
}

// ---------------- edge scatter: one wave per edge ----------------
__global__ void k_scatter(const float* __restrict__ h, const int* __restrict__ src,
                          const int* __restrict__ dst, const float* __restrict__ ew,
                          const float* __restrict__ dinv, float* __restrict__ agg, int E) {
    int e    = (int)((blockIdx.x * (size_t)blockDim.x + threadIdx.x) >> 5);
    int lane = threadIdx.x & 31;
    if (e >= E) return;
    int s = src[e], d = dst[e];
    float norm = dinv[s] * ew[e] * dinv[d];
    // lane covers features [4*lane, 4*lane+3] -> coalesced b128 gather
    const float4* hs = (const float4*)(h + (size_t)s * HID);
    float4 v = hs[lane];
    float* ap = agg + (size_t)d * HID + lane * 4;
    atomicAdd(ap + 0, v.x * norm);
    atomicAdd(ap + 1, v.y * norm);
    atomicAdd(ap + 2, v.z * norm);
    atomicAdd(ap + 3, v.w * norm);
}

// ---------------- finalize: self-loop + bias + relu ----------------
__global__ void k_finalize(const float* __restrict__ agg, const float* __restrict__ h,
                           const float* __restrict__ b, const float* __restrict__ dinv,
                           float* __restrict__ out, int total) {
    int i = blockIdx.x * blockDim.x + threadIdx.x;
    if (i >= total) return;
    int n = i >> 7, f = i & (HID - 1);
    float di = dinv[n];
    float v = agg[i] + h[i] * di * di + b[f];
    out[i] = fmaxf(v, 0.0f);
}

// layer-3 variant: accumulate temporal mean directly
__global__ void k_finalize_acc(const float* __restrict__ agg, const float* __restrict__ h,
                               const float* __restrict__ b, const float* __restrict__ dinv,
                               float* __restrict__ acc, float scale, int total) {
    int i = blockIdx.x * blockDim.x + threadIdx.x;
    if (i >= total) return;
    int n = i >> 7, f = i & (HID - 1);
    float di = dinv[n];
    float v = agg[i] + h[i] * di * di + b[f];
    acc[i] += fmaxf(v, 0.0f) * scale;
}

// ---------------- FC head: [BN,128] @ [128,6] + bias ----------------
__global__ void k_fc(const float* __restrict__ acc, const float* __restrict__ fcW,
                     const float* __restrict__ fcb, float* __restrict__ out, int BN) {
    int i = blockIdx.x * blockDim.x + threadIdx.x;
    if (i >= BN * PRED) return;
    int n = i / PRED, p = i % PRED;
    const float* a = acc + (size_t)n * HID;
    float s = fcb[p];
#pragma unroll 8
    for (int f = 0; f < HID; ++f) s = fmaf(a[f], fcW[f * PRED + p], s);
    out[i] = s;
}

extern "C" void kernel_launch(void* const* d_in, const int* in_sizes, int n_in,
                              void* d_out, int out_size, void* d_ws, size_t ws_size,
                              hipStream_t stream) {
    const float* x_seq = (const float*)d_in[0];
    const int*   eidx  = (const int*)d_in[1];   // [2, E] row-major
    const float* ew    = (const float*)d_in[2];
    const float* W1 = (const float*)d_in[3]; const float* b1 = (const float*)d_in[4];
    const float* W2 = (const float*)d_in[5]; const float* b2 = (const float*)d_in[6];
    const float* W3 = (const float*)d_in[7]; const float* b3 = (const float*)d_in[8];
    const float* fcW = (const float*)d_in[9]; const float* fcb = (const float*)d_in[10];
    float* out = (float*)d_out;

    const int E  = in_sizes[1] / 2;
    const int BN = in_sizes[0] / (T_STEPS * IN_CH);
    const int* src = eidx;
    const int* dst = eidx + E;

    // workspace carve-out (floats)
    float* ws = (float*)d_ws;
    size_t o = 0;
    float* dinv = ws + o; o += ((size_t)BN + 1023) & ~(size_t)1023;
    float* hA   = ws + o; o += (size_t)BN * HID;
    float* hB   = ws + o; o += (size_t)BN * HID;
    float* agg  = ws + o; o += (size_t)BN * HID;
    float* acc  = ws + o; o += (size_t)BN * HID;

    const int total  = BN * HID;
    const int gTiles = BN / 16;       // BN = 40000 -> 2500 row tiles
    const int thr    = 256;

    // graph normalization (depends only on edges): once per call
    k_init_deg<<<(BN + thr - 1) / thr, thr, 0, stream>>>(dinv, BN);
    k_acc_deg<<<(E + thr - 1) / thr, thr, 0, stream>>>(dst, ew, dinv, E);
    k_rsqrt<<<(BN + thr - 1) / thr, thr, 0, stream>>>(dinv, BN);
    hipMemsetAsync(acc, 0, (size_t)total * sizeof(float), stream);

    const int gScat = (E + 7) / 8;               // 8 waves per block, 1 edge/wave
    const int gElem = (total + thr - 1) / thr;

    for (int t = 0; t < T_STEPS; ++t) {
        const float* xt = x_seq + (size_t)t * BN * IN_CH;

        // layer 1: h = relu(gcn(x, W1, b1))
        k_gemm_wmma<<<gTiles, thr, 0, stream>>>(xt, W1, hA, IN_CH);
        hipMemsetAsync(agg, 0, (size_t)total * sizeof(float), stream);
        k_scatter<<<gScat, thr, 0, stream>>>(hA, src, dst, ew, dinv, agg, E);
        k_finalize<<<gElem, thr, 0, stream>>>(agg, hA, b1, dinv, hB, total);

        // layer 2
        k_gemm_wmma<<<gTiles, thr, 0, stream>>>(hB, W2, hA, HID);
        hipMemsetAsync(agg, 0, (size_t)total * sizeof(float), stream);
        k_scatter<<<gScat, thr, 0, stream>>>(hA, src, dst, ew, dinv, agg, E);
        k_finalize<<<gElem, thr, 0, stream>>>(agg, hA, b2, dinv, hB, total);

        // layer 3 (fused temporal-mean accumulation)
        k_gemm_wmma<<<gTiles, thr, 0, stream>>>(hB, W3, hA, HID);
        hipMemsetAsync(agg, 0, (size_t)total * sizeof(float), stream);
        k_scatter<<<gScat, thr, 0, stream>>>(hA, src, dst, ew, dinv, agg, E);
        k_finalize_acc<<<gElem, thr, 0, stream>>>(agg, hA, b3, dinv, acc,
                                                  1.0f / (float)T_STEPS, total);
    }

    k_fc<<<(BN * PRED + thr - 1) / thr, thr, 0, stream>>>(acc, fcW, fcb, out, BN);
}